// MoleculeGNNVN_9586367005254
// MI455X (gfx1250) — compile-verified
//
#include <hip/hip_runtime.h>
#include <hip/hip_bf16.h>

// Problem constants (match reference)
#define NN 131072      // nodes
#define EE 524288      // edges
#define GG 4096        // graphs (32 nodes each, batch is sorted)
#define DD 256         // emb dim
#define LL 5           // layers
#define FE 16          // edge feature dim
#define EPS_BN 1e-5f

typedef __attribute__((ext_vector_type(2))) float v2f;
typedef __attribute__((ext_vector_type(8))) float v8f;
typedef __attribute__((ext_vector_type(4))) unsigned int u32x4;
typedef __attribute__((ext_vector_type(4))) int i32x4;
typedef __attribute__((ext_vector_type(8))) int i32x8;

#if defined(__has_builtin)
#if __has_builtin(__builtin_amdgcn_tensor_load_to_lds)
#define USE_TDM 1
#endif
#endif
#ifndef USE_TDM
#define USE_TDM 0
#endif

#if USE_TDM
// ---------------------------------------------------------------------------
// Issue a TDM 2D tile load: 16 rows x 256 fp32 columns from a row-major
// matrix with row stride `nc` (elements) into LDS at byte offset lds_off.
// Descriptor packing per CDNA5 ISA 8.3/8.4 (D# group0 128b, group1 256b):
//   g0: [1:0]=count=1, [63:32]=lds_addr, [120:64]=global_addr, [127:126]=2
//   g1: [17:16]=data_size(4B), [79:48]=tensor_dim0, [111:80]=tensor_dim1,
//       [127:112]=tile_dim0, [143:128]=tile_dim1, [207:160]=dim0_stride
// Groups 2/3 zero (2-D tile). Wave-level op; EXEC ignored; TENSORcnt tracked.
// ---------------------------------------------------------------------------
static __device__ inline void tdm_load_b_tile(const float* gsrc,
                                              unsigned lds_off, int nc) {
  const unsigned long long ga = (unsigned long long)(size_t)gsrc;
  const unsigned ncu = (unsigned)nc;
  u32x4 g0;
  g0[0] = 1u;                                           // count=1 (valid)
  g0[1] = lds_off;                                      // lds_addr (bytes)
  g0[2] = (unsigned)(ga & 0xffffffffu);                 // global_addr[31:0]
  g0[3] = (unsigned)((ga >> 32) & 0x01ffffffu) | (2u << 30);  // [56:32]|type=2
  i32x8 g1;
  g1[0] = (int)(2u << 16);                              // data_size = 4 bytes
  g1[1] = (int)((ncu & 0xffffu) << 16);                 // tensor_dim0 lo
  g1[2] = (int)((ncu >> 16) | (16u << 16));             // dim0 hi | tensor_dim1=16
  g1[3] = (int)(256u << 16);                            // tile_dim0 = 256
  g1[4] = (int)16u;                                     // tile_dim1 = 16
  g1[5] = (int)ncu;                                     // tensor_dim0_stride lo
  g1[6] = 0;
  g1[7] = 0;
  const i32x4 z4 = {0, 0, 0, 0};
#if __clang_major__ >= 23
  const i32x8 z8 = {0, 0, 0, 0, 0, 0, 0, 0};
  __builtin_amdgcn_tensor_load_to_lds(g0, g1, z4, z4, z8, 0);
#else
  __builtin_amdgcn_tensor_load_to_lds(g0, g1, z4, z4, 0);
#endif
}
#endif  // USE_TDM

// ---------------------------------------------------------------------------
// h_in = h + vn[batch];  agg = 0   (float4 per thread)
// ---------------------------------------------------------------------------
__global__ __launch_bounds__(256)
void k_hin(const float* __restrict__ h, const float* __restrict__ vn,
           const int* __restrict__ batch, float* __restrict__ hin,
           float* __restrict__ agg) {
  int i   = blockIdx.x * 256 + threadIdx.x;   // float4 index; total NN*DD/4
  int row = i >> 6;                           // DD/4 = 64 float4 per row
  int col = (i & 63) << 2;
  int g   = batch[row];
  const float4 hv = *(const float4*)(h  + (size_t)row * DD + col);
  const float4 vv = *(const float4*)(vn + (size_t)g   * DD + col);
  float4 o; o.x = hv.x + vv.x; o.y = hv.y + vv.y; o.z = hv.z + vv.z; o.w = hv.w + vv.w;
  *(float4*)(hin + (size_t)row * DD + col) = o;
  float4 z; z.x = 0.f; z.y = 0.f; z.z = 0.f; z.w = 0.f;
  *(float4*)(agg + (size_t)row * DD + col) = z;
}

// ---------------------------------------------------------------------------
// vn init from vn_emb0 (Embedding(1,D) broadcast to all graphs)
// ---------------------------------------------------------------------------
__global__ __launch_bounds__(256)
void k_vninit(const float* __restrict__ e0, float* __restrict__ vn) {
  int i = blockIdx.x * 256 + threadIdx.x;     // GG*DD total
  vn[i] = e0[i & (DD - 1)];
}

// ---------------------------------------------------------------------------
// Edge message + scatter-add:
//   msg = relu(h_in[src] + edge_attr @ We + be);  agg[dst] += msg
// One wave handles one edge's 256 features (8 per lane), 8 edges per wave.
// We (16x256 fp32 = 16KB) staged in LDS once per block.
// ---------------------------------------------------------------------------
#define EDGES_PER_WAVE 8
__global__ __launch_bounds__(256)
void k_edge(const float* __restrict__ hin, const float* __restrict__ ea,
            const int* __restrict__ src, const int* __restrict__ dst,
            const float* __restrict__ We, const float* __restrict__ be,
            float* __restrict__ agg) {
  __shared__ float sW[FE * DD];
  __shared__ float sb[DD];
  for (int i = threadIdx.x; i < FE * DD; i += 256) sW[i] = We[i];
  sb[threadIdx.x] = be[threadIdx.x];
  __syncthreads();

  const int wave  = threadIdx.x >> 5;
  const int lane  = threadIdx.x & 31;
  const int ebase = (blockIdx.x * 8 + wave) * EDGES_PER_WAVE;
  const int d0    = lane * 8;

  float bb[8];
#pragma unroll
  for (int j = 0; j < 8; j++) bb[j] = sb[d0 + j];

  for (int t = 0; t < EDGES_PER_WAVE; t++) {
    const int e = ebase + t;
    const int s = src[e];
    const int d = dst[e];
    float a[FE];
#pragma unroll
    for (int k = 0; k < FE; k++) a[k] = ea[(size_t)e * FE + k];
    float m[8];
#pragma unroll
    for (int j = 0; j < 8; j++) m[j] = bb[j];
#pragma unroll
    for (int k = 0; k < FE; k++) {
      const float ak = a[k];
#pragma unroll
      for (int j = 0; j < 8; j++) m[j] = fmaf(ak, sW[k * DD + d0 + j], m[j]);
    }
    const float4* hs = (const float4*)(hin + (size_t)s * DD + d0);
    const float4 h0 = hs[0], h1 = hs[1];
    const float hv[8] = {h0.x, h0.y, h0.z, h0.w, h1.x, h1.y, h1.z, h1.w};
    float* ad = agg + (size_t)d * DD + d0;
#pragma unroll
    for (int j = 0; j < 8; j++) {
      const float msg = fmaxf(hv[j] + m[j], 0.f);
      atomicAdd(ad + j, msg);   // global_atomic_add_f32, L2-resident
    }
  }
}

// ---------------------------------------------------------------------------
// Per-graph add-pool over 32 contiguous rows: vt[g] = vn[g] + sum_j hin[g*32+j]
// ---------------------------------------------------------------------------
__global__ __launch_bounds__(256)
void k_pool(const float* __restrict__ hin, const float* __restrict__ vn,
            float* __restrict__ vt) {
  const int g = blockIdx.x, d = threadIdx.x;
  float s = vn[(size_t)g * DD + d];
  const float* base = hin + (size_t)g * 32 * DD + d;
#pragma unroll
  for (int j = 0; j < 32; j++) s += base[j * DD];
  vt[(size_t)g * DD + d] = s;
}

// ---------------------------------------------------------------------------
// Generic fp32 WMMA GEMM:  C = epilogue( (A [+A2]) @ B [+bias] )
// Block tile 64x256, K-step 16, double-buffered LDS, 8 waves arranged 2(M)x4(N);
// each wave computes a 32x64 tile = 2x4 16x16 WMMA tiles (8 v8f accumulators),
// 8 WMMAs per 6 fragment loads per K=4 step, via V_WMMA_F32_16X16X4_F32.
// B tiles staged by the Tensor Data Mover (wave 0 issues tensor_load_to_lds,
// completion via s_wait_tensorcnt) when available; cooperative loads otherwise.
// Fragment layouts per CDNA5 ISA 7.12.2:
//   A 16x4 f32:  lanes 0-15 -> M=lane, v0=K0/v1=K1; lanes 16-31 -> K2/K3
//   B 4x16 f32:  v0: lanes0-15=K0 row, lanes16-31=K2 row; v1: K1/K3
//   C/D 16x16:   VGPR r: lanes0-15 -> M=r, lanes16-31 -> M=r+8
// ---------------------------------------------------------------------------
template <bool HAS_A2, bool HAS_BIAS, bool HAS_BN, bool HAS_RELU, bool HAS_RES,
          bool ACC_C>
__global__ __launch_bounds__(256)
void gemm_wmma(const float* __restrict__ A, const float* __restrict__ A2,
               const float* __restrict__ B, const float* __restrict__ bias,
               const float* __restrict__ bn_m, const float* __restrict__ bn_v,
               const float* __restrict__ bn_g, const float* __restrict__ bn_b,
               const float* __restrict__ Res, float* __restrict__ C,
               int M, int K, int Nc) {
  constexpr int BM = 64, BN = 256, KT = 16;
  constexpr int AS = 20;  // A stage row stride: 8B-aligned even-k pairs,
                          // 20*row mod 64 distinct for rows 0..15 (no conflicts)
  constexpr int A_FLOATS = 2 * BM * AS;   // double-buffered A stage
  constexpr int B_FLOATS = 2 * KT * BN;   // double-buffered B stage
  // single LDS block -> B-stage byte offsets are known constants for the TDM D#
  __shared__ float smem[A_FLOATS + B_FLOATS];

  const int tid  = threadIdx.x;
  const int wave = tid >> 5, lane = tid & 31;
  const int half = lane >> 4, l16 = lane & 15;
  const int wm = wave & 1;     // wave M position (2 waves along M)
  const int wn = wave >> 1;    // wave N position (4 waves along N)
  const int bm0 = blockIdx.y * BM;
  const int bn0 = blockIdx.x * BN;

  v8f acc[2][4];
#pragma unroll
  for (int i = 0; i < 2; i++)
#pragma unroll
    for (int j = 0; j < 4; j++)
#pragma unroll
      for (int r = 0; r < 8; r++) acc[i][j][r] = 0.f;

  // cooperative-load mappings
  const int arow  = tid >> 2;            // 0..63
  const int acol  = (tid & 3) * 4;       // 0,4,8,12
#if !USE_TDM
  const int brow0 = tid >> 6;            // 0..3 (rows brow0 + 4r)
  const int bcol  = (tid & 63) * 4;      // 0..252
#endif

  auto stage = [&](int kk, int buf) {
    // stage A (64x16), fusing optional A+A2
    float4 av = *(const float4*)(A + (size_t)(bm0 + arow) * K + kk + acol);
    if constexpr (HAS_A2) {
      const float4 a2 =
          *(const float4*)(A2 + (size_t)(bm0 + arow) * K + kk + acol);
      av.x += a2.x; av.y += a2.y; av.z += a2.z; av.w += a2.w;
    }
    float* as = &smem[buf * BM * AS + arow * AS + acol];
    as[0] = av.x; as[1] = av.y; as[2] = av.z; as[3] = av.w;
    // stage B (16x256)
#if USE_TDM
    if (wave == 0)
      tdm_load_b_tile(B + (size_t)kk * Nc + bn0,
                      (unsigned)((A_FLOATS + buf * KT * BN) * 4), Nc);
#else
#pragma unroll
    for (int r = 0; r < 4; r++) {
      const int rr = brow0 + r * 4;
      *(float4*)&smem[A_FLOATS + buf * KT * BN + rr * BN + bcol] =
          *(const float4*)(B + (size_t)(kk + rr) * Nc + bn0 + bcol);
    }
    if (kk + 2 * KT < K)   // prefetch K+2 tile of B (global_prefetch_b8)
      __builtin_prefetch(B + (size_t)(kk + 2 * KT + brow0) * Nc + bn0 + bcol,
                         0, 1);
#endif
  };

  stage(0, 0);
#if USE_TDM
  if (wave == 0) __builtin_amdgcn_s_wait_tensorcnt((short)0);
#endif
  __syncthreads();

  int buf = 0;
  for (int kk = 0; kk < K; kk += KT) {
    if (kk + KT < K) stage(kk + KT, buf ^ 1);  // overlap with compute
    const int ab = buf * BM * AS;
    const int bb = A_FLOATS + buf * KT * BN;
#pragma unroll
    for (int k4 = 0; k4 < KT; k4 += 4) {
      const int kb = k4 + half * 2;
      v2f af[2], bf[4];
#pragma unroll
      for (int mi = 0; mi < 2; mi++) {
        const int row = wm * 32 + mi * 16 + l16;
        af[mi].x = smem[ab + row * AS + kb];
        af[mi].y = smem[ab + row * AS + kb + 1];
      }
#pragma unroll
      for (int ni = 0; ni < 4; ni++) {
        const int col = wn * 64 + ni * 16 + l16;
        bf[ni].x = smem[bb + kb * BN + col];
        bf[ni].y = smem[bb + (kb + 1) * BN + col];
      }
#pragma unroll
      for (int mi = 0; mi < 2; mi++)
#pragma unroll
        for (int ni = 0; ni < 4; ni++)
          acc[mi][ni] = __builtin_amdgcn_wmma_f32_16x16x4_f32(
              false, af[mi], false, bf[ni], (short)0, acc[mi][ni], false,
              false);
    }
#if USE_TDM
    if (wave == 0) __builtin_amdgcn_s_wait_tensorcnt((short)0);
#endif
    __syncthreads();
    buf ^= 1;
  }

  // epilogue: bias -> BN(eval) -> relu -> +residual -> (+=C)
#pragma unroll
  for (int mi = 0; mi < 2; mi++) {
#pragma unroll
    for (int ni = 0; ni < 4; ni++) {
      const int gcol = bn0 + wn * 64 + ni * 16 + l16;
      float bv = 0.f;
      if constexpr (HAS_BIAS) bv = bias[gcol];
      float scale = 1.f, shift = 0.f;
      if constexpr (HAS_BN) {
        scale = bn_g[gcol] * rsqrtf(bn_v[gcol] + EPS_BN);
        shift = bn_b[gcol] - bn_m[gcol] * scale;
      }
#pragma unroll
      for (int r = 0; r < 8; r++) {
        const int grow = bm0 + wm * 32 + mi * 16 + half * 8 + r;
        float x = acc[mi][ni][r] + bv;
        if constexpr (HAS_BN) x = x * scale + shift;
        if constexpr (HAS_RELU) x = fmaxf(x, 0.f);
        const size_t idx = (size_t)grow * Nc + gcol;
        if constexpr (HAS_RES) x += Res[idx];
        if constexpr (ACC_C) x += C[idx];
        C[idx] = x;
      }
    }
  }
}

// ---------------------------------------------------------------------------
extern "C" void kernel_launch(void* const* d_in, const int* in_sizes, int n_in,
                              void* d_out, int out_size, void* d_ws,
                              size_t ws_size, hipStream_t stream) {
  const float* x    = (const float*)d_in[0];
  const int*   eidx = (const int*)d_in[1];
  const float* ea   = (const float*)d_in[2];
  const int*   batch= (const int*)d_in[3];
  const float* vne0 = (const float*)d_in[4];
  const float* We   = (const float*)d_in[5];
  const float* be   = (const float*)d_in[6];
  const float* W1   = (const float*)d_in[7];
  const float* b1   = (const float*)d_in[8];
  const float* W2   = (const float*)d_in[9];
  const float* b2   = (const float*)d_in[10];
  const float* bng  = (const float*)d_in[11];
  const float* bnb  = (const float*)d_in[12];
  const float* bnm  = (const float*)d_in[13];
  const float* bnv  = (const float*)d_in[14];
  const float* Wv1  = (const float*)d_in[15];
  const float* bv1  = (const float*)d_in[16];
  const float* vng  = (const float*)d_in[17];
  const float* vnb  = (const float*)d_in[18];
  const float* vnm  = (const float*)d_in[19];
  const float* vnv  = (const float*)d_in[20];
  const float* Wv2  = (const float*)d_in[21];
  const float* bv2  = (const float*)d_in[22];

  const int* src = eidx;        // edge_index[0]
  const int* dst = eidx + EE;   // edge_index[1]

  // workspace layout (~684 MB): h | hin | agg | tmid | vn | vt | zt
  float* ws   = (float*)d_ws;
  float* h    = ws;
  float* hin  = h    + (size_t)NN * DD;
  float* agg  = hin  + (size_t)NN * DD;
  float* tmid = agg  + (size_t)NN * DD;       // NN x 2D
  float* vn   = tmid + (size_t)NN * 2 * DD;
  float* vt   = vn   + (size_t)GG * DD;
  float* zt   = vt   + (size_t)GG * DD;

  k_vninit<<<(GG * DD) / 256, 256, 0, stream>>>(vne0, vn);

  const dim3 blk(256);
  for (int l = 0; l < LL; l++) {
    const float* hsrc = (l == 0) ? x : h;
    k_hin<<<(NN * DD / 4) / 256, blk, 0, stream>>>(hsrc, vn, batch, hin, agg);

    k_edge<<<EE / (8 * EDGES_PER_WAVE), blk, 0, stream>>>(
        hin, ea, src, dst, We + (size_t)l * FE * DD, be + (size_t)l * DD, agg);

    // GEMM1: tmid = relu((hin + agg) @ W1 + b1)   [N x 512]
    gemm_wmma<true, true, false, true, false, false>
        <<<dim3((2 * DD) / 256, NN / 64), blk, 0, stream>>>(
            hin, agg, W1 + (size_t)l * DD * 2 * DD, b1 + (size_t)l * 2 * DD,
            nullptr, nullptr, nullptr, nullptr, nullptr, tmid, NN, DD, 2 * DD);

    // GEMM2: out = BN(tmid @ W2 + b2) [relu if l<4] + hin
    float* outp = (l == LL - 1) ? (float*)d_out : h;
    if (l < LL - 1) {
      gemm_wmma<false, true, true, true, true, false>
          <<<dim3(DD / 256, NN / 64), blk, 0, stream>>>(
              tmid, nullptr, W2 + (size_t)l * 2 * DD * DD, b2 + (size_t)l * DD,
              bnm + (size_t)l * DD, bnv + (size_t)l * DD, bng + (size_t)l * DD,
              bnb + (size_t)l * DD, hin, outp, NN, 2 * DD, DD);
    } else {
      gemm_wmma<false, true, true, false, true, false>
          <<<dim3(DD / 256, NN / 64), blk, 0, stream>>>(
              tmid, nullptr, W2 + (size_t)l * 2 * DD * DD, b2 + (size_t)l * DD,
              bnm + (size_t)l * DD, bnv + (size_t)l * DD, bng + (size_t)l * DD,
              bnb + (size_t)l * DD, hin, outp, NN, 2 * DD, DD);
    }

    if (l < LL - 1) {
      // virtual-node update: vt = pool(hin) + vn; z = relu(BN(vt@Wv1+bv1));
      // vn += z @ Wv2 + bv2
      k_pool<<<GG, blk, 0, stream>>>(hin, vn, vt);
      gemm_wmma<false, true, true, true, false, false>
          <<<dim3(DD / 256, GG / 64), blk, 0, stream>>>(
              vt, nullptr, Wv1 + (size_t)l * DD * DD, bv1 + (size_t)l * DD,
              vnm + (size_t)l * DD, vnv + (size_t)l * DD, vng + (size_t)l * DD,
              vnb + (size_t)l * DD, nullptr, zt, GG, DD, DD);
      gemm_wmma<false, true, false, false, false, true>
          <<<dim3(DD / 256, GG / 64), blk, 0, stream>>>(
              zt, nullptr, Wv2 + (size_t)l * DD * DD, bv2 + (size_t)l * DD,
              nullptr, nullptr, nullptr, nullptr, nullptr, vn, GG, DD, DD);
    }
  }
  (void)in_sizes; (void)n_in; (void)out_size; (void)ws_size;
}